// Net_71210557767875
// MI455X (gfx1250) — compile-verified
//
#include <hip/hip_runtime.h>

typedef __attribute__((ext_vector_type(2))) float v2f;
typedef __attribute__((ext_vector_type(8))) float v8f;

#define NFEAT 256
#define NCLASS 64

// ---------------------------------------------------------------------------
// Degree kernels
// ---------------------------------------------------------------------------
__global__ void deg_init_kernel(float* __restrict__ deg, int n) {
    int i = blockIdx.x * blockDim.x + threadIdx.x;
    if (i < n) deg[i] = 1.0f;  // self-loop contribution
}

__global__ void deg_count_kernel(const int* __restrict__ col, float* __restrict__ deg, int e) {
    int i = blockIdx.x * blockDim.x + threadIdx.x;
    if (i < e) atomicAdd(&deg[col[i]], 1.0f);
}

__global__ void deg_rsqrt_kernel(float* __restrict__ deg, int n) {
    int i = blockIdx.x * blockDim.x + threadIdx.x;
    if (i < n) deg[i] = rsqrtf(deg[i]);  // deg >= 1 always (self loops)
}

// ---------------------------------------------------------------------------
// GEMM1: H[n x 64] = X[n x 256] @ Wg[256 x 64]   (fp32 WMMA 16x16x4)
// One wave computes a 16-row x 64-col stripe: 4 v8f accumulators.
// A-frag layout: lanes 0-15 -> K = {k0, k0+1}; lanes 16-31 -> K = {k0+2, k0+3}
// C/D layout: VGPR r -> M = r (lanes 0-15) / M = 8+r (lanes 16-31); lane&15 -> N
// ---------------------------------------------------------------------------
__global__ __launch_bounds__(256) void gemm1_kernel(const float* __restrict__ X,
                                                    const float* __restrict__ Wg,
                                                    float* __restrict__ H, int n) {
    const int lane = threadIdx.x & 31;
    const int wave = threadIdx.x >> 5;
    const int mtile = blockIdx.x * 8 + wave;
    const int m0 = mtile * 16;
    if (m0 >= n) return;

    const int half = lane >> 4;   // 0 or 1
    const int l15  = lane & 15;
    int m = m0 + l15;
    if (m >= n) m = n - 1;  // clamp (n is a multiple of 16 for this problem)
    const float* __restrict__ xrow = X + (size_t)m * NFEAT;

    v8f acc0 = {}, acc1 = {}, acc2 = {}, acc3 = {};

    for (int k0 = 0; k0 < NFEAT; k0 += 4) {
        const int kb = k0 + half * 2;
        v2f a;
        a.x = xrow[kb];
        a.y = xrow[kb + 1];

        const float* __restrict__ w0 = Wg + (size_t)kb * NCLASS;        // row kb
        const float* __restrict__ w1 = w0 + NCLASS;                     // row kb+1
        v2f b0, b1, b2, b3;
        b0.x = w0[l15];      b0.y = w1[l15];
        b1.x = w0[16 + l15]; b1.y = w1[16 + l15];
        b2.x = w0[32 + l15]; b2.y = w1[32 + l15];
        b3.x = w0[48 + l15]; b3.y = w1[48 + l15];

        acc0 = __builtin_amdgcn_wmma_f32_16x16x4_f32(false, a, false, b0, (short)0, acc0, false, false);
        acc1 = __builtin_amdgcn_wmma_f32_16x16x4_f32(false, a, false, b1, (short)0, acc1, false, false);
        acc2 = __builtin_amdgcn_wmma_f32_16x16x4_f32(false, a, false, b2, (short)0, acc2, false, false);
        acc3 = __builtin_amdgcn_wmma_f32_16x16x4_f32(false, a, false, b3, (short)0, acc3, false, false);
    }

    const int mbase = m0 + half * 8;
#pragma unroll
    for (int r = 0; r < 8; ++r) {
        const int mr = mbase + r;
        if (mr < n) {
            float* __restrict__ hrow = H + (size_t)mr * NCLASS;
            hrow[l15]      = acc0[r];
            hrow[16 + l15] = acc1[r];
            hrow[32 + l15] = acc2[r];
            hrow[48 + l15] = acc3[r];
        }
    }
}

// ---------------------------------------------------------------------------
// agg init: self-loop contribution + bias fold:  agg = h * dinv^2 + b_gcn
// ---------------------------------------------------------------------------
__global__ void agg_init_kernel(const float* __restrict__ H,
                                const float* __restrict__ dinv,
                                const float* __restrict__ bg,
                                float* __restrict__ agg, int n) {
    int i = blockIdx.x * blockDim.x + threadIdx.x;
    int total = n * NCLASS;
    if (i < total) {
        int node = i >> 6;
        int c = i & 63;
        float d = dinv[node];
        agg[i] = H[i] * d * d + bg[c];
    }
}

// ---------------------------------------------------------------------------
// Edge scatter: 16 threads per edge, each does a float4 gather of h[row]
// and 4 global_atomic_add_f32 into agg[col]. h/agg are L2-resident (51 MB).
// ---------------------------------------------------------------------------
__global__ void scatter_kernel(const int* __restrict__ ei,
                               const float* __restrict__ H,
                               const float* __restrict__ dinv,
                               float* __restrict__ agg, int e) {
    int gid = blockIdx.x * blockDim.x + threadIdx.x;
    int total = e * 16;
    if (gid >= total) return;
    const int edge = gid >> 4;
    const int q = gid & 15;
    const int r = ei[edge];       // source (row)
    const int c = ei[e + edge];   // target (col)
    const float nrm = dinv[r] * dinv[c];

    const float4 hv = *reinterpret_cast<const float4*>(H + (size_t)r * NCLASS + q * 4);
    float* __restrict__ dst = agg + (size_t)c * NCLASS + q * 4;
    atomicAdd(dst + 0, hv.x * nrm);
    atomicAdd(dst + 1, hv.y * nrm);
    atomicAdd(dst + 2, hv.z * nrm);
    atomicAdd(dst + 3, hv.w * nrm);
}

// ---------------------------------------------------------------------------
// GEMM2: OUT[n x 64] = relu(agg)[n x 64] @ Wl[64 x 64] + b_lin  (fp32 WMMA)
// ---------------------------------------------------------------------------
__global__ __launch_bounds__(256) void gemm2_kernel(const float* __restrict__ AGG,
                                                    const float* __restrict__ Wl,
                                                    const float* __restrict__ bl,
                                                    float* __restrict__ OUT, int n) {
    const int lane = threadIdx.x & 31;
    const int wave = threadIdx.x >> 5;
    const int mtile = blockIdx.x * 8 + wave;
    const int m0 = mtile * 16;
    if (m0 >= n) return;

    const int half = lane >> 4;
    const int l15  = lane & 15;
    int m = m0 + l15;
    if (m >= n) m = n - 1;
    const float* __restrict__ arow = AGG + (size_t)m * NCLASS;

    v8f acc0 = {}, acc1 = {}, acc2 = {}, acc3 = {};

    for (int k0 = 0; k0 < NCLASS; k0 += 4) {
        const int kb = k0 + half * 2;
        v2f a;
        a.x = fmaxf(arow[kb], 0.0f);       // fused ReLU on A operand
        a.y = fmaxf(arow[kb + 1], 0.0f);

        const float* __restrict__ w0 = Wl + (size_t)kb * NCLASS;
        const float* __restrict__ w1 = w0 + NCLASS;
        v2f b0, b1, b2, b3;
        b0.x = w0[l15];      b0.y = w1[l15];
        b1.x = w0[16 + l15]; b1.y = w1[16 + l15];
        b2.x = w0[32 + l15]; b2.y = w1[32 + l15];
        b3.x = w0[48 + l15]; b3.y = w1[48 + l15];

        acc0 = __builtin_amdgcn_wmma_f32_16x16x4_f32(false, a, false, b0, (short)0, acc0, false, false);
        acc1 = __builtin_amdgcn_wmma_f32_16x16x4_f32(false, a, false, b1, (short)0, acc1, false, false);
        acc2 = __builtin_amdgcn_wmma_f32_16x16x4_f32(false, a, false, b2, (short)0, acc2, false, false);
        acc3 = __builtin_amdgcn_wmma_f32_16x16x4_f32(false, a, false, b3, (short)0, acc3, false, false);
    }

    const float bl0 = bl[l15];
    const float bl1 = bl[16 + l15];
    const float bl2 = bl[32 + l15];
    const float bl3 = bl[48 + l15];

    const int mbase = m0 + half * 8;
#pragma unroll
    for (int r = 0; r < 8; ++r) {
        const int mr = mbase + r;
        if (mr < n) {
            float* __restrict__ orow = OUT + (size_t)mr * NCLASS;
            orow[l15]      = acc0[r] + bl0;
            orow[16 + l15] = acc1[r] + bl1;
            orow[32 + l15] = acc2[r] + bl2;
            orow[48 + l15] = acc3[r] + bl3;
        }
    }
}

// ---------------------------------------------------------------------------
// Launcher
// ---------------------------------------------------------------------------
extern "C" void kernel_launch(void* const* d_in, const int* in_sizes, int n_in,
                              void* d_out, int out_size, void* d_ws, size_t ws_size,
                              hipStream_t stream) {
    const float* x  = (const float*)d_in[0];
    const int*   ei = (const int*)d_in[1];
    const float* Wg = (const float*)d_in[2];
    const float* bg = (const float*)d_in[3];
    const float* Wl = (const float*)d_in[4];
    const float* bl = (const float*)d_in[5];
    float* out = (float*)d_out;

    const int n = in_sizes[0] / NFEAT;   // 100000
    const int e = in_sizes[1] / 2;       // 3200000

    // Workspace layout (floats): [dinv: n][h: n*64][agg: n*64]
    float* ws   = (float*)d_ws;
    size_t off  = ((size_t)n + 3) & ~(size_t)3;  // 16B-align following blocks
    float* dinv = ws;
    float* h    = ws + off;
    float* agg  = h + (size_t)n * NCLASS;

    const int T = 256;

    deg_init_kernel <<<(n + T - 1) / T, T, 0, stream>>>(dinv, n);
    deg_count_kernel<<<(e + T - 1) / T, T, 0, stream>>>(ei + e, dinv, e);
    deg_rsqrt_kernel<<<(n + T - 1) / T, T, 0, stream>>>(dinv, n);

    const int mtiles = (n + 15) / 16;
    const int gblocks = (mtiles + 7) / 8;   // 8 waves per block, 1 tile each
    gemm1_kernel<<<gblocks, T, 0, stream>>>(x, Wg, h, n);

    agg_init_kernel<<<((n * NCLASS) + T - 1) / T, T, 0, stream>>>(h, dinv, bg, agg, n);

    const int stotal = e * 16;
    scatter_kernel<<<(stotal + T - 1) / T, T, 0, stream>>>(ei, h, dinv, agg, e);

    gemm2_kernel<<<gblocks, T, 0, stream>>>(agg, Wl, bl, out, n);
}